// ShrinkTM_9629316677930
// MI455X (gfx1250) — compile-verified
//
#include <hip/hip_runtime.h>
#include <math.h>

typedef __attribute__((ext_vector_type(2))) float v2f;
typedef __attribute__((ext_vector_type(8))) float v8f;

#define NSAMP 64
#define MDIM  30
#define KPAD  32
#define XSTR  34   // padded LDS row stride for xs (keeps float2 8B-aligned, breaks bank conflicts)
#define GSTR  65   // padded LDS row stride for G

__device__ __forceinline__ float nan2num(float v) {
    if (v != v) return 0.0f;                       // NaN -> 0
    if (__builtin_isinf(v))
        return v > 0.f ? 3.4028234663852886e38f : -3.4028234663852886e38f;
    return v;
}

__global__ __launch_bounds__(64) void shrink_loc_kernel(
    const float* __restrict__ aug,      // [NSAMP, nloc, MDIM+1]
    const float* __restrict__ scales,   // [nloc]
    const float* __restrict__ shrink,   // [nloc]
    const float* __restrict__ meanf,    // [nloc, MDIM]
    const float* __restrict__ theta,    // [5]
    float* __restrict__ loglik_out,     // [nloc]
    int nloc)
{
    __shared__ float xs[NSAMP * XSTR];
    __shared__ float G[NSAMP * GSTR];
    __shared__ float r_s[NSAMP];
    __shared__ float sqv[NSAMP];
    __shared__ float scv[KPAD];

    const int tid  = threadIdx.x;
    const int loc  = blockIdx.x;
    const int lane = tid & 31;
    const int wv   = tid >> 5;

    const float th0 = theta[0], th1 = theta[1], th2 = theta[2];
    const float th3 = theta[3], th4 = theta[4];

    // ---- per-feature scaling: scv[m] = (m+1)^(-exp(th4)) / (exp(th3)*sqrt(3)) ----
    if (tid < KPAD) {
        float v = 0.f;
        if (tid < MDIM) {
            float e4 = expf(th4);
            float ls = expf(th3) * 1.7320508075688772f;  // sqrt(2*1.5)
            v = powf((float)(tid + 1), -e4) / ls;
        }
        scv[tid] = v;
    }
    __syncthreads();

    // ---- load row tid (sample tid of this location), scale into LDS, compute rhs ----
    {
        const size_t base = ((size_t)tid * (size_t)nloc + (size_t)loc) * (size_t)(MDIM + 1);
        const float resp = aug[base];
        float fm = 0.f;
        #pragma unroll
        for (int m = 0; m < MDIM; ++m) {
            float v = nan2num(aug[base + 1 + m]);
            fm += v * meanf[(size_t)loc * MDIM + m];
            xs[tid * XSTR + m] = v * scv[m];
        }
        xs[tid * XSTR + MDIM]     = 0.f;   // K padding 30,31
        xs[tid * XSTR + MDIM + 1] = 0.f;
        r_s[tid] = resp - fm;
    }
    __syncthreads();

    // ---- Gram = xs @ xs^T via V_WMMA_F32_16X16X4_F32 (exact f32) ----
    // wave w computes row-blocks {2w, 2w+1} x all 4 col-blocks; 8 K-steps of 4.
    {
        const int rsel = (lane >= 16) ? 1 : 0;
        const int m15  = lane & 15;
        for (int ib = 2 * wv; ib < 2 * wv + 2; ++ib) {
            v2f Aop[8];
            const int arow = ib * 16 + m15;
            #pragma unroll
            for (int kk = 0; kk < 8; ++kk) {
                // A 16x4 layout: lanes0-15 K={4kk,4kk+1}, lanes16-31 K={4kk+2,4kk+3}
                const float* p = &xs[arow * XSTR + 4 * kk + 2 * rsel];
                v2f a; a.x = p[0]; a.y = p[1];
                Aop[kk] = a;
            }
            for (int jb = 0; jb < 4; ++jb) {
                v8f acc = {};
                const int brow = jb * 16 + m15;
                #pragma unroll
                for (int kk = 0; kk < 8; ++kk) {
                    // B 4x16 row-striped: VGPR0 = rows K={4kk,4kk+1}, VGPR1 = K={4kk+2,4kk+3}
                    v2f b;
                    b.x = xs[brow * XSTR + 4 * kk + rsel];
                    b.y = xs[brow * XSTR + 4 * kk + 2 + rsel];
                    acc = __builtin_amdgcn_wmma_f32_16x16x4_f32(
                        false, Aop[kk], false, b, (short)0, acc, false, false);
                }
                // C/D layout: row = 16*ib + v + 8*rsel, col = 16*jb + (lane&15)
                const int rbase = ib * 16 + 8 * rsel;
                const int col   = jb * 16 + m15;
                #pragma unroll
                for (int v = 0; v < 8; ++v) G[(rbase + v) * GSTR + col] = acc[v];
            }
        }
    }
    __syncthreads();

    // ---- snapshot squared norms (diagonal) before overwriting G ----
    sqv[tid] = G[tid * GSTR + tid];
    __syncthreads();

    // ---- Matern-1.5 transform: G = c_loc*(1+s3d)exp(-s3d) + I ; loc 0 -> I ----
    {
        const float sigma = expf(th1 + th2 * logf(scales[loc]));
        const float cloc  = sigma * sigma / shrink[loc];
        const float sqi   = sqv[tid];
        if (loc == 0) {
            for (int j = 0; j < NSAMP; ++j)
                G[tid * GSTR + j] = (tid == j) ? 1.f : 0.f;
        } else {
            for (int j = 0; j < NSAMP; ++j) {
                float dij = G[tid * GSTR + j];
                float d2  = fmaxf(sqi + sqv[j] - 2.f * dij, 0.f);
                float s3d = 1.7320508075688772f * sqrtf(d2);
                float kv  = (1.f + s3d) * expf(-s3d);
                G[tid * GSTR + j] = cloc * kv + ((tid == j) ? 1.f : 0.f);
            }
        }
    }

    // ---- fused right-looking Cholesky + forward solve L y = r ----
    float logdet = 0.f, ss = 0.f;
    for (int k = 0; k < NSAMP; ++k) {
        __syncthreads();                       // trailing/rhs updates of iter k-1 visible
        const float lkk = sqrtf(G[k * GSTR + k]);
        if (tid == 0) logdet += logf(lkk);
        float lik = 0.f;
        if (tid > k) {
            lik = G[tid * GSTR + k] / lkk;
            G[tid * GSTR + k] = lik;
        }
        if (tid == k) G[k * GSTR + k] = lkk;
        __syncthreads();                       // column k finalized
        const float yk = r_s[k] / lkk;
        if (tid == 0) ss += yk * yk;
        if (tid > k) {
            r_s[tid] -= lik * yk;              // forward substitution update
            for (int j = k + 1; j <= tid; ++j) // trailing update, row tid (lower tri)
                G[tid * GSTR + j] -= lik * G[j * GSTR + k];
        }
    }

    // ---- per-location log-likelihood ----
    if (tid == 0) {
        const float nug_mult   = 1.f - 1.f / (expf(th0) + 1.f);
        const float nug_mean   = shrink[loc];
        const float alpha      = 1.f / (nug_mult * nug_mult) + 2.f;
        const float beta       = nug_mean * (alpha - 1.f);
        const float alpha_post = alpha + 0.5f * (float)NSAMP;
        const float beta_post  = beta + 0.5f * ss;
        const float ll = -logdet + alpha * logf(beta) - alpha_post * logf(beta_post)
                         + lgammaf(alpha_post) - lgammaf(alpha);
        loglik_out[loc] = ll;
    }
}

__global__ __launch_bounds__(256) void shrink_reduce_kernel(
    const float* __restrict__ in, float* __restrict__ out, int n)
{
    __shared__ float sm[256];
    float s = 0.f;
    for (int i = threadIdx.x; i < n; i += 256) s += in[i];
    sm[threadIdx.x] = s;
    __syncthreads();
    for (int off = 128; off > 0; off >>= 1) {
        if ((int)threadIdx.x < off) sm[threadIdx.x] += sm[threadIdx.x + off];
        __syncthreads();
    }
    if (threadIdx.x == 0) out[0] = -sm[0];
}

extern "C" void kernel_launch(void* const* d_in, const int* in_sizes, int n_in,
                              void* d_out, int out_size, void* d_ws, size_t ws_size,
                              hipStream_t stream) {
    const float* aug    = (const float*)d_in[0];
    const float* scales = (const float*)d_in[1];
    const float* shrink = (const float*)d_in[2];
    const float* meanf  = (const float*)d_in[3];
    const float* theta  = (const float*)d_in[4];
    const int nloc = in_sizes[1];               // 8192
    float* ws = (float*)d_ws;                   // per-location loglik scratch

    shrink_loc_kernel<<<nloc, 64, 0, stream>>>(aug, scales, shrink, meanf, theta, ws, nloc);
    shrink_reduce_kernel<<<1, 256, 0, stream>>>(ws, (float*)d_out, nloc);
}